// Predict_Masked_Audio_Tokens_86723979641258
// MI455X (gfx1250) — compile-verified
//
#include <hip/hip_runtime.h>

// Problem constants (match the JAX reference)
#define B_   16
#define Q_   4096
#define D_   1024
#define M_   2048
#define OUT_ 32

typedef float v2f __attribute__((ext_vector_type(2)));
typedef float v8f __attribute__((ext_vector_type(8)));

#define KCHUNK          256   // K values of W staged per LDS chunk (32 KB)
#define ROWS_PER_BLOCK  64    // 4 waves x 16 rows
#define THREADS         128

// -------- Kernel 1: zero the whole [B,Q,OUT] output (8 MB) --------
__global__ __launch_bounds__(256) void pmat_zero_kernel(float4* __restrict__ p, int n4) {
    int i = blockIdx.x * blockDim.x + threadIdx.x;
    if (i < n4) p[i] = make_float4(0.f, 0.f, 0.f, 0.f);
}

// -------- Kernel 2: gather -> GEMM(D x 32) via f32 WMMA -> bias -> scatter --------
__global__ __launch_bounds__(THREADS) void pmat_gemm_kernel(
    const float* __restrict__ dA,    // [B, Q, D]
    const int*   __restrict__ midx,  // [B, M]
    const float* __restrict__ W,     // [OUT, D] row-major
    const float* __restrict__ bias,  // [OUT]
    float*       __restrict__ out)   // [B, Q, OUT]
{
    // W^T chunk: ldsW[kpair * 32 + o] = { W[o][kbase+2*kpair], W[o][kbase+2*kpair+1] }
    __shared__ v2f ldsW[(KCHUNK / 2) * OUT_];   // 32 KB

    const int tid   = threadIdx.x;
    const int lane  = tid & 31;
    const int wave  = tid >> 5;
    const int batch = blockIdx.y;
    const int rowTile = blockIdx.x * ROWS_PER_BLOCK + wave * 16;

    const int r    = lane & 15;   // row (A) / col (B) within the 16-wide stripe
    const int half = lane >> 4;   // 0: K offsets {0,1}; 1: K offsets {2,3} per K-step of 4

    // This lane's gathered A row. Lane covers K = {4t+2*half, 4t+2*half+1}.
    const int  aRow = rowTile + r;
    const long long gRow = (long long)midx[batch * M_ + aRow];
    const float* aPtr = dA + ((size_t)batch * Q_ + (size_t)gRow) * (size_t)D_ + 2 * half;

    v8f c0 = {};   // output cols 0..15
    v8f c1 = {};   // output cols 16..31

    for (int kc = 0; kc < D_; kc += KCHUNK) {
        // Stage transposed W chunk into LDS. Consecutive tids -> consecutive kpairs:
        // coalesced 8B global loads; LDS writes are a one-time 32 KB cost.
        for (int i = tid; i < OUT_ * (KCHUNK / 2); i += THREADS) {
            int o     = i / (KCHUNK / 2);
            int kpair = i % (KCHUNK / 2);
            v2f w2 = *(const v2f*)(W + (size_t)o * D_ + kc + 2 * kpair);
            ldsW[kpair * OUT_ + o] = w2;
        }
        __syncthreads();

        #pragma unroll 8
        for (int k = 0; k < KCHUNK; k += 4) {
            // A fragment: 16x4 f32, one aligned 8B load per lane
            v2f a = *(const v2f*)(aPtr + kc + k);
            // B fragments: 4x16 f32 per tile; single ds_load_b64 per tile
            int kp = (k >> 1) + half;
            v2f b0 = ldsW[kp * OUT_ + r];
            v2f b1 = ldsW[kp * OUT_ + 16 + r];
            // D = A x B + C  (exact f32 path: V_WMMA_F32_16X16X4_F32)
            c0 = __builtin_amdgcn_wmma_f32_16x16x4_f32(false, a, false, b0,
                                                       (short)0, c0, false, false);
            c1 = __builtin_amdgcn_wmma_f32_16x16x4_f32(false, a, false, b1,
                                                       (short)0, c1, false, false);
        }
        __syncthreads();
    }

    // Epilogue: bias add + scatter. C/D layout: VGPR v, lane l -> M = v + 8*half, N = r.
    const float bv0 = bias[r];
    const float bv1 = bias[16 + r];

    int orow[8];
    #pragma unroll
    for (int v = 0; v < 8; ++v)
        orow[v] = midx[batch * M_ + rowTile + v + 8 * half];

    #pragma unroll
    for (int v = 0; v < 8; ++v) {
        size_t base = ((size_t)batch * Q_ + (size_t)orow[v]) * (size_t)OUT_;
        out[base + r]      = c0[v] + bv0;
        out[base + 16 + r] = c1[v] + bv1;
    }
}

extern "C" void kernel_launch(void* const* d_in, const int* in_sizes, int n_in,
                              void* d_out, int out_size, void* d_ws, size_t ws_size,
                              hipStream_t stream) {
    const float* dA   = (const float*)d_in[0];   // [B,Q,D] f32
    const int*   midx = (const int*)  d_in[1];   // [B,M]   int
    const float* W    = (const float*)d_in[2];   // [OUT,D] f32
    const float* bias = (const float*)d_in[3];   // [OUT]   f32
    float*       out  = (float*)d_out;           // [B,Q,OUT] f32

    // 1) zero output (unmasked rows must be 0)
    const int n4 = (B_ * Q_ * OUT_) / 4;
    pmat_zero_kernel<<<(n4 + 255) / 256, 256, 0, stream>>>((float4*)out, n4);

    // 2) gather + projection + scatter: 512 workgroups x 4 waves
    dim3 grid(M_ / ROWS_PER_BLOCK, B_);
    pmat_gemm_kernel<<<grid, THREADS, 0, stream>>>(dA, midx, W, bias, out);
}